// layer_2_to_2_anydim_39883066311070
// MI455X (gfx1250) — compile-verified
//
#include <hip/hip_runtime.h>
#include <hip/hip_bf16.h>

typedef __attribute__((ext_vector_type(2))) float v2f;
typedef __attribute__((ext_vector_type(8))) float v8f;

#define N_ 32
#define D_ 128
#define S_ 128
#define M_ 64
#define MM_ (M_ * M_)   // 4096

// ---------------- workspace layout (floats) ----------------
// mc   : [N*D*M]          mean of columns        262144
// dg   : [N*D*M]          diagonal               262144
// md   : [N*D]            mean of diagonal         4096
// ma   : [N*D]            mean of everything       4096
// c2p  : [D/2][S][2]      packed c2 for WMMA A    16384
// Abuf : [N*S*M]          rank-1 correction      262144
// Bbuf : [N*S]            scalar correction        4096
#define OFF_MC   0
#define OFF_DG   (OFF_MC + N_*D_*M_)
#define OFF_MD   (OFF_DG + N_*D_*M_)
#define OFF_MA   (OFF_MD + N_*D_)
#define OFF_C2P  (OFF_MA + N_*D_)
#define OFF_AB   (OFF_C2P + D_*S_)
#define OFF_BB   (OFF_AB + N_*S_*M_)

// ---------------------------------------------------------------------------
// Kernel 1: per-(n,d) statistics.  One 64-thread block per (n,d) pair.
// ---------------------------------------------------------------------------
__global__ __launch_bounds__(64)
void stats_kernel(const float* __restrict__ inp, float* __restrict__ ws) {
    const int nd = blockIdx.x;           // 0 .. N*D-1
    const int j  = threadIdx.x;          // 0 .. 63  (column index)
    const float* p = inp + (size_t)nd * MM_;

    float colsum = 0.0f;
    #pragma unroll 8
    for (int i = 0; i < M_; ++i)
        colsum += p[i * M_ + j];         // coalesced across lanes
    const float mcj = colsum * (1.0f / M_);
    const float dgj = p[j * M_ + j];

    ws[OFF_MC + nd * M_ + j] = mcj;
    ws[OFF_DG + nd * M_ + j] = dgj;

    __shared__ float s1[M_], s2[M_];
    s1[j] = mcj; s2[j] = dgj;
    __syncthreads();
    for (int off = 32; off > 0; off >>= 1) {
        if (j < off) { s1[j] += s1[j + off]; s2[j] += s2[j + off]; }
        __syncthreads();
    }
    if (j == 0) {
        ws[OFF_MD + nd] = s2[0] * (1.0f / M_);   // mean of diagonal
        ws[OFF_MA + nd] = s1[0] * (1.0f / M_);   // mean over all elements
    }
}

// ---------------------------------------------------------------------------
// Kernel 2: pack c2 = coeffs[:,:,1] as c2p[(d>>1)*2S + s*2 + (d&1)]
// so a WMMA A-fragment is one contiguous 64-bit load per lane.
// ---------------------------------------------------------------------------
__global__ __launch_bounds__(256)
void pack_kernel(const float* __restrict__ coeffs, float* __restrict__ ws) {
    const int t = blockIdx.x * 256 + threadIdx.x;   // 0 .. D*S-1
    if (t >= D_ * S_) return;
    const int d = t / S_;
    const int s = t % S_;
    ws[OFF_C2P + (d >> 1) * (2 * S_) + s * 2 + (d & 1)] =
        coeffs[(size_t)(d * S_ + s) * 5 + 1];
}

// ---------------------------------------------------------------------------
// Kernel 3: A[n,s,i] = sum_d 0.5*(c1[d,s]*mc[n,d,i] + c3[d,s]*dg[n,d,i])
//           B[n,s]   = sum_d (c4[d,s]*md[n,d] + c5[d,s]*ma[n,d])
// Tiny, fully L2-resident.
// ---------------------------------------------------------------------------
__global__ __launch_bounds__(256)
void ab_kernel(const float* __restrict__ coeffs, float* __restrict__ ws) {
    const int t = blockIdx.x * 256 + threadIdx.x;   // 0 .. N*S*M-1
    const int n = t >> 13;            // / (S*M)
    const int r = t & 8191;
    const int s = r >> 6;             // / M
    const int i = r & 63;

    const float* mc = ws + OFF_MC + (size_t)n * D_ * M_ + i;
    const float* dg = ws + OFF_DG + (size_t)n * D_ * M_ + i;
    float acc = 0.0f;
    #pragma unroll 4
    for (int d = 0; d < D_; ++d) {
        const float c1 = coeffs[(size_t)(d * S_ + s) * 5 + 0];
        const float c3 = coeffs[(size_t)(d * S_ + s) * 5 + 2];
        acc += 0.5f * (c1 * mc[d * M_] + c3 * dg[d * M_]);
    }
    ws[OFF_AB + (size_t)(n * S_ + s) * M_ + i] = acc;

    if (i == 0) {
        const float* md = ws + OFF_MD + n * D_;
        const float* ma = ws + OFF_MA + n * D_;
        float bb = 0.0f;
        #pragma unroll 4
        for (int d = 0; d < D_; ++d) {
            bb += coeffs[(size_t)(d * S_ + s) * 5 + 3] * md[d];
            bb += coeffs[(size_t)(d * S_ + s) * 5 + 4] * ma[d];
        }
        ws[OFF_BB + n * S_ + s] = bb;
    }
}

// ---------------------------------------------------------------------------
// Kernel 4: main GEMM.  out2[n, s, ij] = sum_d c2[d,s] * inp[n, d, ij]
// One wave per 16x16 tile of (s, ij).  K-loop: 32 x V_WMMA_F32_16X16X4_F32.
// Epilogue fuses A[n,s,i] + A[n,s,j] + B[n,s] + bias[s].
//
// A-fragment (16x4, M=s):  lanes 0-15 hold {K=k0,k0+1}, lanes 16-31 {k0+2,k0+3}
// B-fragment (4x16, N=ij): lanes 0-15 hold rows {k0,k0+1}, lanes 16-31 {k0+2,k0+3}
// C/D (16x16): VGPR v: lanes 0-15 -> row s0+v, lanes 16-31 -> row s0+v+8
// ---------------------------------------------------------------------------
__global__ __launch_bounds__(256)
void gemm_kernel(const float* __restrict__ inp,
                 const float* __restrict__ bias,
                 const float* __restrict__ ws,
                 float* __restrict__ out) {
    const int lane = threadIdx.x & 31;
    const int wv   = blockIdx.x * 8 + (threadIdx.x >> 5);   // global wave id
    const int ijt  = wv & 255;          // 256 tiles over M*M
    const int st   = (wv >> 8) & 7;     // 8 tiles over S
    const int n    = wv >> 11;          // 32

    const int half = lane >> 4;         // 0: lanes 0-15, 1: lanes 16-31
    const int lh   = lane & 15;

    const int s_lane  = st * 16 + lh;       // A row for this lane
    const int ij_lane = ijt * 16 + lh;      // B column for this lane

    const float* inpN = inp + (size_t)n * D_ * MM_;
    const float* c2p  = ws + OFF_C2P;

    v8f acc = {};
    #pragma unroll 4
    for (int k0 = 0; k0 < D_; k0 += 4) {
        const int ka = k0 + 2 * half;       // this half-wave's K base
        const v2f a = *(const v2f*)(c2p + (ka >> 1) * (2 * S_) + s_lane * 2);
        v2f b;
        b.x = inpN[(size_t)ka * MM_ + ij_lane];
        b.y = inpN[(size_t)(ka + 1) * MM_ + ij_lane];
        acc = __builtin_amdgcn_wmma_f32_16x16x4_f32(
            /*neg_a=*/false, a, /*neg_b=*/false, b,
            /*c_mod=*/(short)0, acc, /*reuse_a=*/false, /*reuse_b=*/false);
    }

    // epilogue: 16 | 64, so all 16 columns of this tile share the same i
    const int i  = (ijt * 16) >> 6;         // row index in [0,64)
    const int jc = ij_lane & 63;            // column index for this lane

    const float* Ab = ws + OFF_AB;
    const float* Bb = ws + OFF_BB;
    const float Aj = Ab[(size_t)(n * S_ + st * 16) * M_ + 0];  // dummy warm ref (optimized out)
    (void)Aj;

    #pragma unroll
    for (int v = 0; v < 8; ++v) {
        const int s = st * 16 + v + 8 * half;
        const size_t row = (size_t)(n * S_ + s);
        const float add = Ab[row * M_ + i] + Ab[row * M_ + jc]
                        + Bb[row] + bias[s];
        out[(row * M_ + i) * M_ + jc] = acc[v] + add;
    }
}

// ---------------------------------------------------------------------------
extern "C" void kernel_launch(void* const* d_in, const int* in_sizes, int n_in,
                              void* d_out, int out_size, void* d_ws, size_t ws_size,
                              hipStream_t stream) {
    const float* inp    = (const float*)d_in[0];   // [N, D, M, M]
    const float* coeffs = (const float*)d_in[1];   // [D, S, 5]
    const float* bias   = (const float*)d_in[2];   // [1, S, 1, 1]
    float* out = (float*)d_out;                    // [N, S, M, M]
    float* ws  = (float*)d_ws;

    // 1) per-(n,d) statistics
    stats_kernel<<<N_ * D_, 64, 0, stream>>>(inp, ws);

    // 2) pack c2 for WMMA A-fragments
    pack_kernel<<<(D_ * S_ + 255) / 256, 256, 0, stream>>>(coeffs, ws);

    // 3) rank-1 / scalar corrections
    ab_kernel<<<(N_ * S_ * M_) / 256, 256, 0, stream>>>(coeffs, ws);

    // 4) main WMMA GEMM + fused epilogue
    // waves = N * (S/16) * (MM/16) = 32*8*256 = 65536 ; 8 waves per block
    gemm_kernel<<<65536 / 8, 256, 0, stream>>>(inp, bias, ws, out);
}